// TinyQuantumLayer_36842229465702
// MI455X (gfx1250) — compile-verified
//
#include <hip/hip_runtime.h>
#include <math.h>

typedef __attribute__((ext_vector_type(2))) float v2f;
typedef __attribute__((ext_vector_type(8))) float v8f;

#define NQ       10
#define DIM      1024
#define NLAYERS  4
#define WPB      8          // waves (samples) per block
#define PI_F     3.14159265358979323846f

struct c2 { float re, im; };
__device__ __forceinline__ c2 cmul(c2 a, c2 b) {
    return { a.re * b.re - a.im * b.im, a.re * b.im + a.im * b.re };
}
__device__ __forceinline__ c2 cadd(c2 a, c2 b) { return { a.re + b.re, a.im + b.im }; }

// ---------------------------------------------------------------------------
// Prep: fuse RX(t0)*RY(t1)*RZ(t2) per (layer,qubit) into a 4x4 real matrix M
// such that [Re a0, Im a0, Re a1, Im a1] * M applies the gate (row-vector
// convention matching einsum('bmi,ij->bmj')).
// ---------------------------------------------------------------------------
__global__ void qprep_kernel(const float* __restrict__ theta, float* __restrict__ Mout) {
    int g = blockIdx.x * blockDim.x + threadIdx.x;
    if (g >= NLAYERS * NQ) return;
    float t0 = theta[g * 3 + 0], t1 = theta[g * 3 + 1], t2 = theta[g * 3 + 2];
    float cx = cosf(t0 * 0.5f), sx = sinf(t0 * 0.5f);
    float cy = cosf(t1 * 0.5f), sy = sinf(t1 * 0.5f);
    float cz = cosf(t2 * 0.5f), sz = sinf(t2 * 0.5f);
    c2 X[2][2] = { { {cx, 0.f}, {0.f, -sx} }, { {0.f, -sx}, {cx, 0.f} } };
    c2 Y[2][2] = { { {cy, 0.f}, {-sy, 0.f} }, { {sy, 0.f}, {cy, 0.f} } };
    c2 Z[2][2] = { { {cz, -sz}, {0.f, 0.f} }, { {0.f, 0.f}, {cz, sz} } };
    c2 XY[2][2], G[2][2];
    for (int i = 0; i < 2; ++i)
        for (int j = 0; j < 2; ++j)
            XY[i][j] = cadd(cmul(X[i][0], Y[0][j]), cmul(X[i][1], Y[1][j]));
    for (int i = 0; i < 2; ++i)
        for (int j = 0; j < 2; ++j)
            G[i][j] = cadd(cmul(XY[i][0], Z[0][j]), cmul(XY[i][1], Z[1][j]));
    float* M = Mout + g * 16;
    for (int i = 0; i < 2; ++i)
        for (int j = 0; j < 2; ++j) {
            M[(2 * i    ) * 4 + 2 * j    ] =  G[i][j].re;
            M[(2 * i    ) * 4 + 2 * j + 1] =  G[i][j].im;
            M[(2 * i + 1) * 4 + 2 * j    ] = -G[i][j].im;
            M[(2 * i + 1) * 4 + 2 * j + 1] =  G[i][j].re;
        }
}

// ---------------------------------------------------------------------------
// Main: one wave32 per sample; 1024-amplitude state lives in LDS (8KB/sample).
// Gates applied with V_WMMA_F32_16X16X4_F32 (16 pairs x K=4 per issue).
// Index map factorization:  i0(blk*16 + t) = base(blk) + off(t), where
// base(blk) is wave-uniform (SALU) and off(t) is loop-invariant per lane.
// ---------------------------------------------------------------------------
__global__ void __launch_bounds__(WPB * 32)
qsim_kernel(const float* __restrict__ x, const float* __restrict__ encW,
            const float* __restrict__ encB, const float* __restrict__ Mvar,
            float* __restrict__ out, int batch) {
    __shared__ float2 lds[WPB * DIM];

    const int lane = threadIdx.x & 31;
    const int w    = threadIdx.x >> 5;
    const int b    = blockIdx.x * WPB + w;
    if (b >= batch) return;

    float2* st = &lds[w * DIM];
    float*  stf = (float*)st;
    const int n  = lane & 15;   // N column / pair-row (for A)
    const int hi = lane >> 4;   // lane-half: A amplitude half, B K-half

    // |0...0>
    for (int i = lane; i < DIM; i += 32) st[i] = make_float2(0.f, 0.f);
    if (lane == 0) st[0] = make_float2(1.f, 0.f);

    // Apply one single-qubit gate at bit position s = 9-q; Bop holds the 4x4
    // real gate matrix in columns 0..3 of the 4x16 B operand.
    auto apply = [&](int s, v2f Bop) {
        const int lowmask = (1 << s) - 1;
        auto off = [&](int t) { return ((t >> s) << (s + 1)) | (t & lowmask); };

        // loop-invariant per-lane float indices
        const int ldIdx = (off(n) + (hi << s)) * 2;              // ds_load_b64
        const int half = n >> 1, ri = n & 1;
        int sIdx[8];
#pragma unroll
        for (int r = 0; r < 8; ++r)
            sIdx[r] = (off(r + 8 * hi) + (half << s)) * 2 + ri;  // ds_store_b32
        const bool act = (n < 4);

        // software-pipelined over 32 pair-blocks; base(blk) is uniform
        int bb = 0;                                   // base(0) = 0
        float2 a = *(const float2*)&stf[ldIdx];
        for (int blk = 0; blk < 32; ++blk) {
            int nb = bb;
            float2 an = a;
            if (blk < 31) {
                int u = (blk + 1) << 4;
                nb = ((((u >> s) << (s + 1)) | (u & lowmask))) * 2;
                an = *(const float2*)&stf[ldIdx + nb];           // prefetch next
            }
            v2f A; A.x = a.x; A.y = a.y;              // K = 2*hi, 2*hi+1
            v8f C = {0.f, 0.f, 0.f, 0.f, 0.f, 0.f, 0.f, 0.f};
            v8f D = __builtin_amdgcn_wmma_f32_16x16x4_f32(
                        false, A, false, Bop, (short)0, C, false, false);
            if (act) {                                 // cols 0..3 hold result
#pragma unroll
                for (int r = 0; r < 8; ++r) stf[sIdx[r] + bb] = D[r];
            }
            a = an; bb = nb;
        }
    };

    // ---- encoding RY(enc[b][q]) ----
    float xv0 = x[b * 3 + 0], xv1 = x[b * 3 + 1], xv2 = x[b * 3 + 2];
    for (int q = 0; q < NQ; ++q) {
        float ang = PI_F * tanhf(xv0 * encW[q * 3] + xv1 * encW[q * 3 + 1] +
                                 xv2 * encW[q * 3 + 2] + encB[q]);
        float c = cosf(ang * 0.5f), sv = sinf(ang * 0.5f);
        // RY 4x4: M[k][n] = c if n==k; (k<2 ? -s : s) if n==k^2; else 0
        int k0 = 2 * hi, k1 = 2 * hi + 1;
        v2f Bop;
        Bop.x = (n == k0) ? c : ((n == (k0 ^ 2)) ? ((k0 < 2) ? -sv : sv) : 0.f);
        Bop.y = (n == k1) ? c : ((n == (k1 ^ 2)) ? ((k1 < 2) ? -sv : sv) : 0.f);
        apply(9 - q, Bop);
    }

    // ---- variational layers ----
    for (int layer = 0; layer < NLAYERS; ++layer) {
        for (int q = 0; q < NQ; ++q) {
            const float* M = Mvar + (layer * NQ + q) * 16;
            v2f Bop;
            Bop.x = (n < 4) ? M[(2 * hi    ) * 4 + n] : 0.f;
            Bop.y = (n < 4) ? M[(2 * hi + 1) * 4 + n] : 0.f;
            apply(9 - q, Bop);
        }
        // CNOT ring (q -> q+1 mod 10): swap ctrl=1,tgt=0 <-> ctrl=1,tgt=1
        for (int q = 0; q < NQ; ++q) {
            int cpos = 9 - q, tpos = 9 - ((q + 1) % NQ);
            int lo = cpos < tpos ? cpos : tpos;
            int hp = cpos < tpos ? tpos : cpos;
            for (int k = lane; k < 256; k += 32) {
                int v = ((k >> lo) << (lo + 1)) | (k & ((1 << lo) - 1));
                v     = ((v >> hp) << (hp + 1)) | (v & ((1 << hp) - 1));
                int i0 = v | (1 << cpos);
                int i1 = i0 | (1 << tpos);
                float2 a0 = st[i0];
                float2 a1 = st[i1];
                st[i0] = a1;
                st[i1] = a0;
            }
        }
    }

    // ---- Z expectations: out[b][q] = sum_i |amp_i|^2 * (1 - 2*bit(i,q)) ----
    float acc[NQ];
#pragma unroll
    for (int q = 0; q < NQ; ++q) acc[q] = 0.f;
    for (int i = lane; i < DIM; i += 32) {
        float2 a = st[i];
        float pr = a.x * a.x + a.y * a.y;
#pragma unroll
        for (int q = 0; q < NQ; ++q) {
            unsigned bit = ((unsigned)i >> (9 - q)) & 1u;
            acc[q] += __int_as_float(__float_as_int(pr) ^ (bit << 31));
        }
    }
#pragma unroll
    for (int off = 16; off > 0; off >>= 1)
#pragma unroll
        for (int q = 0; q < NQ; ++q)
            acc[q] += __shfl_xor(acc[q], off, 32);
    if (lane == 0) {
#pragma unroll
        for (int q = 0; q < NQ; ++q) out[b * NQ + q] = acc[q];
    }
}

// ---------------------------------------------------------------------------
extern "C" void kernel_launch(void* const* d_in, const int* in_sizes, int n_in,
                              void* d_out, int out_size, void* d_ws, size_t ws_size,
                              hipStream_t stream) {
    const float* x     = (const float*)d_in[0];   // [batch, 3]
    const float* encW  = (const float*)d_in[1];   // [10, 3]
    const float* encB  = (const float*)d_in[2];   // [10]
    const float* theta = (const float*)d_in[3];   // [4, 10, 3]
    float* out  = (float*)d_out;                  // [batch, 10]
    float* Mvar = (float*)d_ws;                   // 40 * 16 floats

    int batch = in_sizes[0] / 3;

    qprep_kernel<<<1, 64, 0, stream>>>(theta, Mvar);

    int blocks = (batch + WPB - 1) / WPB;
    qsim_kernel<<<blocks, WPB * 32, 0, stream>>>(x, encW, encB, Mvar, out, batch);
}